// GCN_82334523064536
// MI455X (gfx1250) — compile-verified
//
#include <hip/hip_runtime.h>

#define NN 50000
#define NN_PAD 50016   // round up to 32-row GEMM tile grid; only xw is padded
#define NE 800000
#define FF 64
#define NCLS 8
#define NGR 500

typedef __attribute__((ext_vector_type(2))) float v2f;
typedef __attribute__((ext_vector_type(8))) float v8f;

__global__ void fillf(float* __restrict__ p, int n, float v) {
    int i = blockIdx.x * blockDim.x + threadIdx.x;
    if (i < n) p[i] = v;
}

// degree over targets (edges + self loops)
__global__ void degree_kernel(const int* __restrict__ col, float* __restrict__ deg,
                              int E, int Etot) {
    int e = blockIdx.x * blockDim.x + threadIdx.x;
    if (e >= Etot) return;
    int c = (e < E) ? col[e] : (e - E);
    atomicAdd(&deg[c], 1.0f);
}

__global__ void rsqrt_kernel(const float* __restrict__ deg, float* __restrict__ dinv, int n) {
    int i = blockIdx.x * blockDim.x + threadIdx.x;
    if (i >= n) return;
    float d = deg[i];
    dinv[i] = (d > 0.0f) ? rsqrtf(d) : 0.0f;
}

// Y[NN_PAD,64] = X[n,64] @ W[64,64] using V_WMMA_F32_16X16X4_F32 (exact fp32).
// Block = 256 threads = 8 waves; wave w -> rowTile = w/4 (2 per block), colTile = w%4.
// Y is padded to NN_PAD rows so stores are unconditional (EXEC stays all-ones).
__global__ __launch_bounds__(256) void gemm64_wmma(const float* __restrict__ X,
                                                   const float* __restrict__ W,
                                                   float* __restrict__ Y, int n) {
    __shared__ float sW[64 * 64];
    int tid = threadIdx.x;
#pragma unroll
    for (int i = 0; i < 16; ++i) sW[tid + 256 * i] = W[tid + 256 * i];
    __syncthreads();

    int wave = tid >> 5;
    int ln   = tid & 31;
    int half = ln >> 4;      // which 16-lane half
    int l16  = ln & 15;
    int colT = wave & 3;
    int m0   = blockIdx.x * 32 + (wave >> 2) * 16;

    int m = m0 + l16;                 // A row for this lane (all lanes M=l16)
    if (m >= n) m = n - 1;            // clamp ragged-tail loads; output rows are padded
    int ncol = colT * 16 + l16;       // B/D column for this lane

    v8f acc = {};
#pragma unroll
    for (int kk = 0; kk < 16; ++kk) {
        int k = kk * 4 + 2 * half;    // A: VGPR0=K(+0), VGPR1=K(+1); halves offset by 2
        v2f a = *(const v2f*)(X + (size_t)m * 64 + k);
        v2f b;                         // B: VGPRv holds row K=k+v for this half
        b.x = sW[k * 64 + ncol];
        b.y = sW[(k + 1) * 64 + ncol];
        acc = __builtin_amdgcn_wmma_f32_16x16x4_f32(false, a, false, b,
                                                    (short)0, acc, false, false);
    }
    // D: VGPRv = M=v (lanes<16) / M=v+8 (lanes>=16); unconditional, one base +
    // immediate row offsets (rows are 256B apart -> 24-bit ioffset).
    float* dst = Y + ((size_t)(m0 + 8 * half)) * 64 + ncol;
#pragma unroll
    for (int v = 0; v < 8; ++v) dst[(size_t)v * 64] = acc[v];
}

// 16 lanes per edge, float4 per lane: gather xw[row], scale by norm, atomic-add to out[col]
__global__ void scatter_kernel(const float* __restrict__ xw, const float* __restrict__ dinv,
                               const int* __restrict__ row, const int* __restrict__ col,
                               float* __restrict__ out, int E, int Etot) {
    long long w = (long long)blockIdx.x * blockDim.x + threadIdx.x;
    int e  = (int)(w >> 4);
    int c4 = ((int)w & 15) << 2;
    if (e >= Etot) return;
    int r, c;
    float nrm;
    if (e < E) {
        r = row[e]; c = col[e];
        nrm = dinv[r] * dinv[c];
    } else {
        r = c = e - E;
        float dv = dinv[r];
        nrm = dv * dv;
    }
    float4 v = *(const float4*)(xw + (size_t)r * 64 + c4);
    float* dst = out + (size_t)c * 64 + c4;
    atomicAdd(dst + 0, v.x * nrm);
    atomicAdd(dst + 1, v.y * nrm);
    atomicAdd(dst + 2, v.z * nrm);
    atomicAdd(dst + 3, v.w * nrm);
}

__global__ void bias_act(float* __restrict__ h, const float* __restrict__ b,
                         int n64, int relu) {
    int i = blockIdx.x * blockDim.x + threadIdx.x;
    if (i >= n64) return;
    float v = h[i] + b[i & 63];
    h[i] = (relu && v < 0.0f) ? 0.0f : v;
}

__global__ void pool_accum(const float* __restrict__ h, const int* __restrict__ batch,
                           float* __restrict__ sums, float* __restrict__ cnts, int n) {
    int i = blockIdx.x * blockDim.x + threadIdx.x;
    int node = i >> 6, f = i & 63;
    if (node >= n) return;
    int g = batch[node];
    atomicAdd(&sums[g * 64 + f], h[i]);
    if (f == 0) atomicAdd(&cnts[g], 1.0f);
}

__global__ void final_linear(const float* __restrict__ sums, const float* __restrict__ cnts,
                             const float* __restrict__ Wl, const float* __restrict__ bl,
                             float* __restrict__ out) {
    int i = blockIdx.x * blockDim.x + threadIdx.x;
    if (i >= NGR * NCLS) return;
    int g = i >> 3, c = i & 7;
    float inv = 1.0f / fmaxf(cnts[g], 1.0f);
    float acc = bl[c];
#pragma unroll
    for (int k = 0; k < 64; ++k)
        acc += sums[g * 64 + k] * inv * Wl[k * NCLS + c];
    out[i] = acc;
}

extern "C" void kernel_launch(void* const* d_in, const int* in_sizes, int n_in,
                              void* d_out, int out_size, void* d_ws, size_t ws_size,
                              hipStream_t stream) {
    const float* x    = (const float*)d_in[0];
    const int*   edge = (const int*)d_in[1];
    const int*   row  = edge;            // edge_index[0]
    const int*   col  = edge + NE;       // edge_index[1]
    const int*   batch = (const int*)d_in[2];
    const float* W1 = (const float*)d_in[3];
    const float* b1 = (const float*)d_in[4];
    const float* W2 = (const float*)d_in[5];
    const float* b2 = (const float*)d_in[6];
    const float* W3 = (const float*)d_in[7];
    const float* b3 = (const float*)d_in[8];
    const float* Wl = (const float*)d_in[9];
    const float* bl = (const float*)d_in[10];
    float* out = (float*)d_out;

    float* deg  = (float*)d_ws;                     // NN
    float* dinv = deg + NN;                         // NN
    float* xw   = dinv + NN;                        // NN_PAD*64 (padded GEMM dest)
    float* h    = xw + (size_t)NN_PAD * 64;         // NN*64
    float* sums = h + (size_t)NN * 64;              // NGR*64
    float* cnts = sums + NGR * 64;                  // NGR

    const int Etot = NE + NN;
    const int TB = 256;

    fillf<<<(NN + TB - 1) / TB, TB, 0, stream>>>(deg, NN, 0.0f);
    degree_kernel<<<(Etot + TB - 1) / TB, TB, 0, stream>>>(col, deg, NE, Etot);
    rsqrt_kernel<<<(NN + TB - 1) / TB, TB, 0, stream>>>(deg, dinv, NN);

    const float* Ws[3] = {W1, W2, W3};
    const float* bs[3] = {b1, b2, b3};
    const float* in = x;
    for (int layer = 0; layer < 3; ++layer) {
        gemm64_wmma<<<NN_PAD / 32, TB, 0, stream>>>(in, Ws[layer], xw, NN);
        fillf<<<(NN * 64 + TB - 1) / TB, TB, 0, stream>>>(h, NN * 64, 0.0f);
        long long sw = (long long)Etot * 16;
        scatter_kernel<<<(unsigned)((sw + TB - 1) / TB), TB, 0, stream>>>(
            xw, dinv, row, col, h, NE, Etot);
        bias_act<<<(NN * 64 + TB - 1) / TB, TB, 0, stream>>>(h, bs[layer], NN * 64,
                                                             layer < 2 ? 1 : 0);
        in = h;
    }

    fillf<<<(NGR * 64 + TB - 1) / TB, TB, 0, stream>>>(sums, NGR * 64, 0.0f);
    fillf<<<(NGR + TB - 1) / TB, TB, 0, stream>>>(cnts, NGR, 0.0f);
    pool_accum<<<(NN * 64 + TB - 1) / TB, TB, 0, stream>>>(h, batch, sums, cnts, NN);
    final_linear<<<(NGR * NCLS + TB - 1) / TB, TB, 0, stream>>>(sums, cnts, Wl, bl, out);
}